// AtomGNN_56169582297457
// MI455X (gfx1250) — compile-verified
//
#include <hip/hip_runtime.h>

// AtomGNN for MI455X (gfx1250, wave32).
//
// Heavy edge MLP runs on V_WMMA_F32_16X16X4_F32 (full fp32, matches reference
// precision). One wave processes a 16-edge tile:
//   layer1: [16 x 68] @ [68 x 32]  -> 17 K-steps x 2 N-tiles = 34 WMMAs
//   layer2: [16 x 32] @ [32 x 32]  ->  8 K-steps x 2 N-tiles = 16 WMMAs
// VGPR layouts assumed per CDNA5 ISA 7.12.2:
//   A (16x4 f32): lane L -> row M=L%16, vgpr v + 2*(L/16) -> K
//   B (4x16 f32): lane L -> col N=L%16, vgpr v + 2*(L/16) -> K (mirror of A)
//   C/D (16x16):  lane L, vgpr v -> M = v + 8*(L/16), N = L%16

#define HID 32
#define K1  68          // 2*HID + N_EDGE_FEATURES
#define EPT 16          // edges per wave tile
#define YSTRIDE 34      // padded LDS row stride (even, bank-conflict free)

typedef __attribute__((ext_vector_type(2))) float v2f;
typedef __attribute__((ext_vector_type(8))) float v8f;

__device__ __forceinline__ v8f wmma4(v2f a, v2f b, v8f c) {
  // (neg_a, A, neg_b, B, c_mod, C, reuse_a, reuse_b)
  return __builtin_amdgcn_wmma_f32_16x16x4_f32(false, a, false, b, (short)0, c,
                                               false, false);
}

// ---------------------------------------------------------------- encoder ---
__global__ void __launch_bounds__(256)
enc_kernel(const float* __restrict__ x, const float* __restrict__ w1,
           const float* __restrict__ b1, const float* __restrict__ w2,
           const float* __restrict__ b2, float* __restrict__ h, int n) {
  __shared__ float sw1[16 * HID], sw2[HID * HID], sb1[HID], sb2[HID];
  for (int i = threadIdx.x; i < 16 * HID; i += blockDim.x) sw1[i] = w1[i];
  for (int i = threadIdx.x; i < HID * HID; i += blockDim.x) sw2[i] = w2[i];
  if (threadIdx.x < HID) {
    sb1[threadIdx.x] = b1[threadIdx.x];
    sb2[threadIdx.x] = b2[threadIdx.x];
  }
  __syncthreads();
  int i = blockIdx.x * blockDim.x + threadIdx.x;
  if (i >= n) return;
  float xi[16];
#pragma unroll
  for (int k = 0; k < 16; ++k) xi[k] = x[(long)i * 16 + k];
  float t[HID];
  for (int j = 0; j < HID; ++j) {
    float s = sb1[j];
#pragma unroll
    for (int k = 0; k < 16; ++k) s += xi[k] * sw1[k * HID + j];
    t[j] = fmaxf(s, 0.0f);
  }
  for (int j = 0; j < HID; ++j) {
    float s = sb2[j];
#pragma unroll 8
    for (int k = 0; k < HID; ++k) s += t[k] * sw2[k * HID + j];
    h[(long)i * HID + j] = s;
  }
}

// ---------------------------------------------------- edge message kernel ---
__global__ void __launch_bounds__(256)
edge_kernel(const float* __restrict__ h, const int* __restrict__ edges,
            const float* __restrict__ ef, const float* __restrict__ w1,
            const float* __restrict__ b1, const float* __restrict__ w2,
            const float* __restrict__ b2, float* __restrict__ agg,
            long long n_edges, long long n_tiles) {
  __shared__ float sW1[K1 * HID];
  __shared__ float sW2[HID * HID];
  __shared__ float sB1[HID], sB2[HID];
  __shared__ __align__(16) float sY[8][16 * YSTRIDE];  // per-wave transpose tile

  for (int i = threadIdx.x; i < K1 * HID; i += blockDim.x) sW1[i] = w1[i];
  for (int i = threadIdx.x; i < HID * HID; i += blockDim.x) sW2[i] = w2[i];
  if (threadIdx.x < HID) {
    sB1[threadIdx.x] = b1[threadIdx.x];
    sB2[threadIdx.x] = b2[threadIdx.x];
  }
  __syncthreads();

  const int wave = threadIdx.x >> 5;
  const int lane = threadIdx.x & 31;
  const int half = lane >> 4;   // which K pair / which M half
  const int m    = lane & 15;   // A row within tile; also N column index
  const int col  = m;
  const int ko   = 2 * half;

  // Loop-invariant B fragments + biases cached in VGPRs.
  v2f bw1[17][2], bw2[8][2];
#pragma unroll
  for (int t = 0; t < 17; ++t) {
    const int kb = 4 * t + ko;
#pragma unroll
    for (int nt = 0; nt < 2; ++nt) {
      v2f b;
      b.x = sW1[kb * HID + nt * 16 + col];
      b.y = sW1[(kb + 1) * HID + nt * 16 + col];
      bw1[t][nt] = b;
    }
  }
#pragma unroll
  for (int t = 0; t < 8; ++t) {
    const int kb = 4 * t + ko;
#pragma unroll
    for (int nt = 0; nt < 2; ++nt) {
      v2f b;
      b.x = sW2[kb * HID + nt * 16 + col];
      b.y = sW2[(kb + 1) * HID + nt * 16 + col];
      bw2[t][nt] = b;
    }
  }
  const float bias1_0 = sB1[col], bias1_1 = sB1[16 + col];
  const float bias2_0 = sB2[col], bias2_1 = sB2[16 + col];
  float* myY = &sY[wave][0];

  const long long tile0   = (long long)blockIdx.x * 8 + wave;
  const long long tstride = (long long)gridDim.x * 8;

  for (long long tile = tile0; tile < n_tiles; tile += tstride) {
    const long long ebase = tile * EPT;
    long long e = ebase + m;
    if (e >= n_edges) e = n_edges - 1;  // clamp: keep EXEC all-ones for WMMA
    const int srcn = edges[2 * e];
    const int dstn = edges[2 * e + 1];
    const float* hs = h + (long long)srcn * HID;
    const float* hd = h + (long long)dstn * HID;

    // Gather A fragments: lane holds K pair (4t+2*half, +1) of its edge row.
    v2f a[17];
#pragma unroll
    for (int t = 0; t < 8; ++t) a[t] = *(const v2f*)(hs + 4 * t + ko);
#pragma unroll
    for (int t = 0; t < 8; ++t) a[8 + t] = *(const v2f*)(hd + 4 * t + ko);
    a[16] = *(const v2f*)(ef + e * 4 + ko);

    // Layer 1: y = relu(in @ W1 + b1), K = 68.
    v8f acc0 = {}, acc1 = {};
#pragma unroll
    for (int t = 0; t < 17; ++t) {
      acc0 = wmma4(a[t], bw1[t][0], acc0);
      acc1 = wmma4(a[t], bw1[t][1], acc1);
    }

    // D-layout -> A-layout transpose through per-wave LDS tile (padded).
#pragma unroll
    for (int v = 0; v < 8; ++v) {
      const int row = v + 8 * half;
      myY[row * YSTRIDE + col]      = fmaxf(acc0[v] + bias1_0, 0.0f);
      myY[row * YSTRIDE + 16 + col] = fmaxf(acc1[v] + bias1_1, 0.0f);
    }
    asm volatile("" ::: "memory");  // same-wave LDS ops are in-order

    // Layer 2: z = y @ W2, K = 32.
    v8f z0 = {}, z1 = {};
#pragma unroll
    for (int t = 0; t < 8; ++t) {
      v2f ya = *(const v2f*)(myY + m * YSTRIDE + 4 * t + ko);
      z0 = wmma4(ya, bw2[t][0], z0);
      z1 = wmma4(ya, bw2[t][1], z1);
    }
    asm volatile("" ::: "memory");  // reads done before next-iter overwrite

    // Scatter m = z + b2 into agg[dst] with f32 atomics (coalesced per row).
#pragma unroll
    for (int v = 0; v < 8; ++v) {
      const int row = v + 8 * half;          // M index of this D register
      const int rdst = __shfl(dstn, row, 32);  // dst of edge (ebase+row)
      if (ebase + row < n_edges) {
        float* p = agg + (long long)rdst * HID;
        atomicAdd(p + col, z0[v] + bias2_0);
        atomicAdd(p + 16 + col, z1[v] + bias2_1);
      }
    }
  }
}

// ------------------------------------------------------- node update MLP ---
__global__ void __launch_bounds__(256)
upd_kernel(float* __restrict__ h, const float* __restrict__ agg,
           const float* __restrict__ w1, const float* __restrict__ b1,
           const float* __restrict__ w2, const float* __restrict__ b2, int n) {
  __shared__ float sw1[2 * HID * HID], sw2[HID * HID], sb1[HID], sb2[HID];
  for (int i = threadIdx.x; i < 2 * HID * HID; i += blockDim.x) sw1[i] = w1[i];
  for (int i = threadIdx.x; i < HID * HID; i += blockDim.x) sw2[i] = w2[i];
  if (threadIdx.x < HID) {
    sb1[threadIdx.x] = b1[threadIdx.x];
    sb2[threadIdx.x] = b2[threadIdx.x];
  }
  __syncthreads();
  int i = blockIdx.x * blockDim.x + threadIdx.x;
  if (i >= n) return;
  float xi[2 * HID];
#pragma unroll 8
  for (int k = 0; k < HID; ++k) {
    xi[k] = h[(long)i * HID + k];
    xi[HID + k] = agg[(long)i * HID + k];
  }
  float t[HID];
  for (int j = 0; j < HID; ++j) {
    float s = sb1[j];
#pragma unroll 8
    for (int k = 0; k < 2 * HID; ++k) s += xi[k] * sw1[k * HID + j];
    t[j] = fmaxf(s, 0.0f);
  }
  for (int j = 0; j < HID; ++j) {
    float s = sb2[j];
#pragma unroll 8
    for (int k = 0; k < HID; ++k) s += t[k] * sw2[k * HID + j];
    h[(long)i * HID + j] = xi[j] + s;  // residual
  }
}

// ----------------------------------------------------------------- head ----
__global__ void __launch_bounds__(256)
head_kernel(const float* __restrict__ h, const float* __restrict__ w1,
            const float* __restrict__ b1, const float* __restrict__ w2,
            const float* __restrict__ b2, float* __restrict__ out, int n) {
  __shared__ float sw1[HID * HID], sw2[HID], sb1[HID];
  __shared__ float sb2s;
  for (int i = threadIdx.x; i < HID * HID; i += blockDim.x) sw1[i] = w1[i];
  if (threadIdx.x < HID) {
    sw2[threadIdx.x] = w2[threadIdx.x];
    sb1[threadIdx.x] = b1[threadIdx.x];
  }
  if (threadIdx.x == 0) sb2s = b2[0];
  __syncthreads();
  int i = blockIdx.x * blockDim.x + threadIdx.x;
  if (i >= n) return;
  float xi[HID];
#pragma unroll 8
  for (int k = 0; k < HID; ++k) xi[k] = h[(long)i * HID + k];
  float s2 = sb2s;
  for (int j = 0; j < HID; ++j) {
    float s = sb1[j];
#pragma unroll 8
    for (int k = 0; k < HID; ++k) s += xi[k] * sw1[k * HID + j];
    s2 += fmaxf(s, 0.0f) * sw2[j];
  }
  out[i] = s2;
}

// ---------------------------------------------------------------- launch ---
extern "C" void kernel_launch(void* const* d_in, const int* in_sizes, int n_in,
                              void* d_out, int out_size, void* d_ws,
                              size_t ws_size, hipStream_t stream) {
  const float* node_features = (const float*)d_in[0];
  const int* edges = (const int*)d_in[1];  // int32 per harness convention
  const float* ef = (const float*)d_in[2];
  const float* enc_w1 = (const float*)d_in[3];
  const float* enc_b1 = (const float*)d_in[4];
  const float* enc_w2 = (const float*)d_in[5];
  const float* enc_b2 = (const float*)d_in[6];
  const float* msg_w1 = (const float*)d_in[7];
  const float* msg_b1 = (const float*)d_in[8];
  const float* msg_w2 = (const float*)d_in[9];
  const float* msg_b2 = (const float*)d_in[10];
  const float* upd_w1 = (const float*)d_in[11];
  const float* upd_b1 = (const float*)d_in[12];
  const float* upd_w2 = (const float*)d_in[13];
  const float* upd_b2 = (const float*)d_in[14];
  const float* head_w1 = (const float*)d_in[15];
  const float* head_b1 = (const float*)d_in[16];
  const float* head_w2 = (const float*)d_in[17];
  const float* head_b2 = (const float*)d_in[18];

  const int n_nodes = in_sizes[0] / 16;           // N_NODE_FEATURES = 16
  const long long n_edges = (long long)in_sizes[2] / 4;  // N_EDGE_FEATURES = 4

  float* h = (float*)d_ws;                         // [n_nodes][32]
  float* agg = h + (size_t)n_nodes * HID;          // [n_nodes][32]

  const int nblk = (n_nodes + 255) / 256;
  enc_kernel<<<nblk, 256, 0, stream>>>(node_features, enc_w1, enc_b1, enc_w2,
                                       enc_b2, h, n_nodes);

  const long long n_tiles = (n_edges + EPT - 1) / EPT;
  int eblocks = (int)((n_tiles + 7) / 8);
  if (eblocks > 4096) eblocks = 4096;  // grid-stride: amortize B fragments

  for (int r = 0; r < 2; ++r) {
    hipMemsetAsync(agg, 0, (size_t)n_nodes * HID * sizeof(float), stream);
    edge_kernel<<<eblocks, 256, 0, stream>>>(
        h, edges, ef, msg_w1 + (size_t)r * K1 * HID, msg_b1 + (size_t)r * HID,
        msg_w2 + (size_t)r * HID * HID, msg_b2 + (size_t)r * HID, agg, n_edges,
        n_tiles);
    upd_kernel<<<nblk, 256, 0, stream>>>(
        h, agg, upd_w1 + (size_t)r * 2 * HID * HID, upd_b1 + (size_t)r * HID,
        upd_w2 + (size_t)r * HID * HID, upd_b2 + (size_t)r * HID, n_nodes);
  }

  head_kernel<<<nblk, 256, 0, stream>>>(h, head_w1, head_b1, head_w2, head_b2,
                                        (float*)d_out, n_nodes);
}